// GAT_30270929502483
// MI455X (gfx1250) — compile-verified
//
#include <hip/hip_runtime.h>
#include <hip/hip_bf16.h>
#include <math.h>

// ---------------------------------------------------------------------------
// 2-layer GAT for MI455X (gfx1250, wave32).
// GEMMs via V_WMMA_F32_16X16X4_F32 (fp32 in/out, no precision loss).
// Edge phase: wave-per-edge gather/scatter with f32 atomics (L2-resident).
// ---------------------------------------------------------------------------

typedef __attribute__((ext_vector_type(2))) float v2f;
typedef __attribute__((ext_vector_type(8))) float v8f;

#define GAT_SLOPE 0.2f

// ---- float atomic max via int-ordered trick (single atomic, no CAS loop) ----
__device__ __forceinline__ void atomicMaxF(float* addr, float val) {
  int iv = __float_as_int(val);
  if (iv >= 0) {
    atomicMax((int*)addr, iv);               // nonneg floats order like ints
  } else {
    atomicMin((unsigned int*)addr, (unsigned int)iv); // neg floats order inversely as uint
  }
}

// ---- edge endpoints with implicit self-loops appended at the end -----------
__device__ __forceinline__ void edge_ends(const long long* __restrict__ ei,
                                          int e, int E_, int& s, int& d) {
  if (e < E_) {
    s = (int)ei[e];                 // edge_index[0][e]
    d = (int)ei[(size_t)E_ + e];    // edge_index[1][e]
  } else {
    s = d = e - E_;                 // self loop
  }
}

// ---------------------------------------------------------------------------
// WMMA GEMM: C[M x NC] = A[M x K] * B[K x NC].  M % 16 == 0, NC % 64 == 0,
// K % 4 == 0.  blockDim = 128 (4 waves); each wave computes one 16x16 tile.
// A-frag (16x4):  lane<16 -> rows m, K = k+0 (v.x), k+1 (v.y)
//                 lane>=16 -> rows m, K = k+2 (v.x), k+3 (v.y)
// B-frag (4x16):  lane n = lane&15; v.x = B[k+2*hi][n], v.y = B[k+2*hi+1][n]
// D (16x16):      vgpr i, lane l -> row i + 8*(l>=16), col l&15
// ---------------------------------------------------------------------------
__global__ void wmma_gemm_f32(const float* __restrict__ A,
                              const float* __restrict__ B,
                              float* __restrict__ C, int K, int NC) {
  const int lane = threadIdx.x & 31;
  const int wave = threadIdx.x >> 5;
  const int m    = lane & 15;
  const int hi   = lane >> 4;
  const int row0 = blockIdx.x * 16;
  const int col0 = (blockIdx.y * 4 + wave) * 16;

  const float* Arow = A + (size_t)(row0 + m) * K;
  v8f acc = {};
  for (int k = 0; k < K; k += 4) {
    const int ka = k + 2 * hi;
    v2f a, b;
    a.x = Arow[ka];
    a.y = Arow[ka + 1];
    b.x = B[(size_t)ka * NC + col0 + m];
    b.y = B[(size_t)(ka + 1) * NC + col0 + m];
    acc = __builtin_amdgcn_wmma_f32_16x16x4_f32(false, a, false, b,
                                                (short)0, acc, false, false);
  }
#pragma unroll
  for (int i = 0; i < 8; ++i) {
    C[(size_t)(row0 + 8 * hi + i) * NC + col0 + m] = acc[i];
  }
}

// ---------------------------------------------------------------------------
// Per-node attention dots: a_s[n,h] = dot(xp[n, h*64:(h+1)*64], att_s[h]).
// One wave per node; lane covers W/32 contiguous channels.
// ---------------------------------------------------------------------------
template <int H>
__global__ void att_dots(const float* __restrict__ xp,
                         const float* __restrict__ att_s,
                         const float* __restrict__ att_d,
                         float* __restrict__ a_s, float* __restrict__ a_d,
                         int Nn) {
  const int lane = threadIdx.x & 31;
  const int node = blockIdx.x * (blockDim.x >> 5) + (threadIdx.x >> 5);
  if (node >= Nn) return;
  constexpr int W  = H * 64;
  constexpr int CH = W / 32;   // 4 (H=2) or 2 (H=1)
  const float* row = xp + (size_t)node * W;
  const int base = lane * CH;
  float ss = 0.f, sd = 0.f;
#pragma unroll
  for (int j = 0; j < CH; ++j) {
    float v = row[base + j];
    ss += v * att_s[base + j];
    sd += v * att_d[base + j];
  }
  // reduce within head group (16 lanes for H=2, 32 lanes for H=1)
  constexpr int MAXM = (H == 2) ? 8 : 16;
#pragma unroll
  for (int msk = MAXM; msk >= 1; msk >>= 1) {
    ss += __shfl_xor(ss, msk, 32);
    sd += __shfl_xor(sd, msk, 32);
  }
  if (lane == 0)              { a_s[(size_t)node * H]     = ss; a_d[(size_t)node * H]     = sd; }
  if (H == 2 && lane == 16)   { a_s[(size_t)node * H + 1] = ss; a_d[(size_t)node * H + 1] = sd; }
}

// ---------------------------------------------------------------------------
// Pass 1 over edges: alpha = leaky_relu(a_s[src]+a_d[dst]); segment max -> m.
// ---------------------------------------------------------------------------
template <int H>
__global__ void edge_alpha_max(const long long* __restrict__ ei,
                               const float* __restrict__ a_s,
                               const float* __restrict__ a_d,
                               float* __restrict__ mbuf, int E_, int Etot) {
  const int e = blockIdx.x * blockDim.x + threadIdx.x;
  if (e >= Etot) return;
  int s, d;
  edge_ends(ei, e, E_, s, d);
#pragma unroll
  for (int h = 0; h < H; ++h) {
    float v = a_s[(size_t)s * H + h] + a_d[(size_t)d * H + h];
    v = (v > 0.f) ? v : GAT_SLOPE * v;
    atomicMaxF(&mbuf[(size_t)d * H + h], v);
  }
}

// ---------------------------------------------------------------------------
// Pass 2: e = exp(alpha - m[dst]); cache e, segment-sum -> z.
// ---------------------------------------------------------------------------
template <int H>
__global__ void edge_exp_sum(const long long* __restrict__ ei,
                             const float* __restrict__ a_s,
                             const float* __restrict__ a_d,
                             const float* __restrict__ mbuf,
                             float* __restrict__ ebuf,
                             float* __restrict__ z, int E_, int Etot) {
  const int e = blockIdx.x * blockDim.x + threadIdx.x;
  if (e >= Etot) return;
  int s, d;
  edge_ends(ei, e, E_, s, d);
#pragma unroll
  for (int h = 0; h < H; ++h) {
    float v = a_s[(size_t)s * H + h] + a_d[(size_t)d * H + h];
    v = (v > 0.f) ? v : GAT_SLOPE * v;
    float ex = expf(v - mbuf[(size_t)d * H + h]);
    ebuf[(size_t)e * H + h] = ex;
    atomicAdd(&z[(size_t)d * H + h], ex);
  }
}

// ---------------------------------------------------------------------------
// Pass 3: agg[dst] += (e/z[dst]) * xp[src].  One wave per edge; lane holds
// H*2 contiguous channels (float4 for H=2, float2 for H=1).
// ---------------------------------------------------------------------------
template <int H>
__global__ void edge_aggregate(const long long* __restrict__ ei,
                               const float* __restrict__ ebuf,
                               const float* __restrict__ z,
                               const float* __restrict__ xp,
                               float* __restrict__ agg, int E_, int Etot) {
  const int lane = threadIdx.x & 31;
  const int e = blockIdx.x * (blockDim.x >> 5) + (threadIdx.x >> 5);
  if (e >= Etot) return;
  int s, d;
  edge_ends(ei, e, E_, s, d);
  constexpr int W  = H * 64;
  constexpr int CH = W / 32;
  const int base = lane * CH;
  const int head = base >> 6;
  const float coef = ebuf[(size_t)e * H + head] / z[(size_t)d * H + head];
  const float* xr = xp + (size_t)s * W + base;
  float* ar = agg + (size_t)d * W + base;
#pragma unroll
  for (int j = 0; j < CH; ++j) atomicAdd(&ar[j], coef * xr[j]);
}

// ---------------------------------------------------------------------------
// Elementwise helpers
// ---------------------------------------------------------------------------
__global__ void fill_f32(float* __restrict__ p, float v, size_t n) {
  size_t i = (size_t)blockIdx.x * blockDim.x + threadIdx.x;
  if (i < n) p[i] = v;
}

// out[n*W + c] = bias[c]   (W power of 2)
__global__ void init_bias(float* __restrict__ out, const float* __restrict__ b,
                          size_t n, int Wmask) {
  size_t i = (size_t)blockIdx.x * blockDim.x + threadIdx.x;
  if (i < n) out[i] = b[i & Wmask];
}

// buf = relu(buf + bias[c])
__global__ void bias_relu(float* __restrict__ buf, const float* __restrict__ b,
                          size_t n, int Wmask) {
  size_t i = (size_t)blockIdx.x * blockDim.x + threadIdx.x;
  if (i >= n) return;
  float v = buf[i] + b[(int)(i & Wmask)];
  buf[i] = (v > 0.f) ? v : 0.f;
}

// ---------------------------------------------------------------------------
// Host-side orchestration (graph-capture safe: no malloc/sync, all on stream)
// ---------------------------------------------------------------------------
extern "C" void kernel_launch(void* const* d_in, const int* in_sizes, int n_in,
                              void* d_out, int out_size, void* d_ws, size_t ws_size,
                              hipStream_t stream) {
  const float*     x        = (const float*)d_in[0];
  const long long* ei       = (const long long*)d_in[1];   // int64 [2,E]
  const float*     W1       = (const float*)d_in[2];       // [128,128]
  const float*     att_src1 = (const float*)d_in[3];       // [2,64]
  const float*     att_dst1 = (const float*)d_in[4];
  const float*     b1       = (const float*)d_in[5];       // [128]
  const float*     W2       = (const float*)d_in[6];       // [128,64]
  const float*     att_src2 = (const float*)d_in[7];       // [1,64]
  const float*     att_dst2 = (const float*)d_in[8];
  const float*     b2       = (const float*)d_in[9];       // [64]
  float*           out      = (float*)d_out;               // [N,64]

  const int Nn   = in_sizes[0] / 128;   // 50000
  const int E_   = in_sizes[1] / 2;     // 1600000
  const int Etot = E_ + Nn;             // + self loops

  // ---- workspace layout (floats) ----
  float* ws = (float*)d_ws;
  size_t off = 0;
  float* xp1 = ws + off; off += (size_t)Nn * 128;
  float* as1 = ws + off; off += (size_t)Nn * 2;
  float* ad1 = ws + off; off += (size_t)Nn * 2;
  float* m1  = ws + off; off += (size_t)Nn * 2;
  float* z1  = ws + off; off += (size_t)Nn * 2;
  float* e1  = ws + off; off += (size_t)Etot * 2;
  float* hb  = ws + off; off += (size_t)Nn * 128;  // agg1, then relu'd h in place
  float* xp2 = ws + off; off += (size_t)Nn * 64;
  float* as2 = ws + off; off += (size_t)Nn;
  float* ad2 = ws + off; off += (size_t)Nn;
  float* m2  = ws + off; off += (size_t)Nn;
  float* z2  = ws + off; off += (size_t)Nn;
  float* e2  = ws + off; off += (size_t)Etot;

  const float NEG_INF = -__builtin_huge_valf();
  auto blocks = [](size_t n, int bs) { return (unsigned)((n + bs - 1) / bs); };

  // ================= Layer 1 (H=2, C=64) =================
  // xp1 = x @ W1
  wmma_gemm_f32<<<dim3(Nn / 16, 128 / 64), 128, 0, stream>>>(x, W1, xp1, 128, 128);
  // attention logits per node
  att_dots<2><<<blocks(Nn, 8), 256, 0, stream>>>(xp1, att_src1, att_dst1, as1, ad1, Nn);
  // init segment buffers
  fill_f32<<<blocks((size_t)Nn * 2, 256), 256, 0, stream>>>(m1, NEG_INF, (size_t)Nn * 2);
  fill_f32<<<blocks((size_t)Nn * 2, 256), 256, 0, stream>>>(z1, 0.f, (size_t)Nn * 2);
  fill_f32<<<blocks((size_t)Nn * 128, 256), 256, 0, stream>>>(hb, 0.f, (size_t)Nn * 128);
  // softmax over destination neighborhoods + aggregate
  edge_alpha_max<2><<<blocks(Etot, 256), 256, 0, stream>>>(ei, as1, ad1, m1, E_, Etot);
  edge_exp_sum<2><<<blocks(Etot, 256), 256, 0, stream>>>(ei, as1, ad1, m1, e1, z1, E_, Etot);
  edge_aggregate<2><<<blocks(Etot, 8), 256, 0, stream>>>(ei, e1, z1, xp1, hb, E_, Etot);
  // h = relu(agg + b1)
  bias_relu<<<blocks((size_t)Nn * 128, 256), 256, 0, stream>>>(hb, b1, (size_t)Nn * 128, 127);

  // ================= Layer 2 (H=1, C=64) =================
  wmma_gemm_f32<<<dim3(Nn / 16, 64 / 64), 128, 0, stream>>>(hb, W2, xp2, 128, 64);
  att_dots<1><<<blocks(Nn, 8), 256, 0, stream>>>(xp2, att_src2, att_dst2, as2, ad2, Nn);
  fill_f32<<<blocks(Nn, 256), 256, 0, stream>>>(m2, NEG_INF, (size_t)Nn);
  fill_f32<<<blocks(Nn, 256), 256, 0, stream>>>(z2, 0.f, (size_t)Nn);
  init_bias<<<blocks((size_t)Nn * 64, 256), 256, 0, stream>>>(out, b2, (size_t)Nn * 64, 63);
  edge_alpha_max<1><<<blocks(Etot, 256), 256, 0, stream>>>(ei, as2, ad2, m2, E_, Etot);
  edge_exp_sum<1><<<blocks(Etot, 256), 256, 0, stream>>>(ei, as2, ad2, m2, e2, z2, E_, Etot);
  edge_aggregate<1><<<blocks(Etot, 8), 256, 0, stream>>>(ei, e2, z2, xp2, out, E_, Etot);
}